// KP_Decoder_3513283248430
// MI455X (gfx1250) — compile-verified
//
#include <hip/hip_runtime.h>

typedef __attribute__((ext_vector_type(2))) float v2f;
typedef __attribute__((ext_vector_type(8))) float v8f;

#define NB 32
#define NG 500
#define NP 500
#define NPP 512   // padded P rows for kall / vT
#define ND 128
#define NH 8
#define NQK 16
#define NE 8
#define NFF 512
#define NT 16000   // NB*NG tokens
#define INV_SQRT_EMB 0.08838834764831845f
#define NEG_INF (-__builtin_inff())

static __device__ inline v8f v8_splat(float x) {
  v8f v;
#pragma unroll
  for (int i = 0; i < 8; ++i) v[i] = x;
  return v;
}

static __device__ inline v8f wmma4(v2f a, v2f b, v8f c) {
  // V_WMMA_F32_16X16X4_F32 : D = A(16x4) * B(4x16) + C(16x16), fp32
  return __builtin_amdgcn_wmma_f32_16x16x4_f32(false, a, false, b, (short)0, c,
                                               false, false);
}

static __device__ inline v2f ld2(const float* p) {
  return *(const v2f*)p;  // 8B-aligned pair load -> global_load_b64 / ds_load_b64
}

// ---------------------------------------------------------------------------
// 0) batched transpose: dst[c*rows + r] = src[r*cols + c]
// ---------------------------------------------------------------------------
__global__ __launch_bounds__(256) void transpose_batched(
    const float* __restrict__ src, float* __restrict__ dst, int rows,
    int cols) {
  const size_t n = (size_t)rows * cols;
  const size_t base = (size_t)blockIdx.y * n;
  const size_t idx = (size_t)blockIdx.x * 256 + threadIdx.x;
  if (idx < n) {
    const int r = (int)(idx / cols), c = (int)(idx % cols);
    dst[base + (size_t)c * rows + r] = src[base + idx];
  }
}

// ---------------------------------------------------------------------------
// 1) K / V projections (column-major weights WkT/WvT):
//    kall  [B, 512, 128]           (rows >= NP hold duplicate data, masked later)
//    vT    [B, H, 16, 512]         (PV-matmul friendly; pads hold duplicates)
//    A fragment in registers (fully-unrolled k loop => no scratch spill);
//    8 n-tiles per block, n-tile loop kept rolled.
// ---------------------------------------------------------------------------
__global__ __launch_bounds__(32) void proj_kv_kernel(
    const float* __restrict__ enc, const float* __restrict__ WkT,
    const float* __restrict__ WvT, float* __restrict__ kall,
    float* __restrict__ vT) {
  const int lane = threadIdx.x & 31;
  const int half = lane >> 4, nl = lane & 15;
  const int pt = blockIdx.x;
  const int b = blockIdx.y >> 1, sel = blockIdx.y & 1;
  const float* WT = sel ? WvT : WkT;
  const int p0 = pt * 16;
  int prow = p0 + nl;
  if (prow >= NP) prow = NP - 1;  // clamp loads; pad rows get duplicate data
  const float* arow = enc + ((size_t)b * NP + prow) * ND;
  v2f areg[32];
#pragma unroll
  for (int i = 0; i < 32; ++i) areg[i] = ld2(arow + i * 4 + half * 2);

#pragma unroll 1
  for (int nt = 0; nt < 8; ++nt) {
    const int n0 = nt * 16;
    const float* wrow = WT + (size_t)(n0 + nl) * 128;
    v8f acc = v8_splat(0.f);
#pragma unroll
    for (int i = 0; i < 32; ++i) {  // FULL unroll: areg indices constant
      v2f bb = ld2(wrow + i * 4 + half * 2);
      acc = wmma4(areg[i], bb, acc);
    }
    if (sel == 0) {
#pragma unroll
      for (int r = 0; r < 8; ++r)
        kall[((size_t)b * NPP + p0 + r + half * 8) * 128 + n0 + nl] = acc[r];
    } else {
      float* vrow = vT + (((size_t)b * NH + nt) * 16 + nl) * NPP;
#pragma unroll
      for (int r = 0; r < 8; ++r) vrow[p0 + r + half * 8] = acc[r];
    }
  }
}

// ---------------------------------------------------------------------------
// 2) qbase[b,n] = graph[b,:] @ Wq[0:128,n]  (graph is broadcast over G)
// ---------------------------------------------------------------------------
__global__ __launch_bounds__(128) void qbase_kernel(
    const float* __restrict__ graph, const float* __restrict__ Wq,
    float* __restrict__ qbase) {
  const int b = blockIdx.x, n = threadIdx.x;
  float s = 0.f;
#pragma unroll 8
  for (int d = 0; d < ND; ++d)
    s += graph[(size_t)b * ND + d] * Wq[(size_t)d * 128 + n];
  qbase[(size_t)b * 128 + n] = s;
}

// ---------------------------------------------------------------------------
// 3) Attention: one wave per (b, h, 16-row g-tile)
// ---------------------------------------------------------------------------
__global__ __launch_bounds__(32) void attn_kernel(
    const float* __restrict__ qbase, const float* __restrict__ Wq,
    const float* __restrict__ capacity, const float* __restrict__ mask,
    const float* __restrict__ kall, const float* __restrict__ vT,
    float* __restrict__ outc) {
  __shared__ float s[16][512];
  const int lane = threadIdx.x;
  const int half = lane >> 4, nl = lane & 15;
  const int gt = blockIdx.x, h = blockIdx.y, b = blockIdx.z;
  const int g0 = gt * 16;
  int grow = g0 + nl;
  if (grow >= NG) grow = NG - 1;  // clamp
  const float cap = capacity[(size_t)b * NG + grow];

  // per-lane q fragment: q = qbase + cap * Wq[capacity row]
  v2f qv[4];
#pragma unroll
  for (int i = 0; i < 4; ++i) {
    const int k = i * 4 + half * 2;
    v2f q01 = ld2(qbase + (size_t)b * 128 + h * 16 + k);
    v2f w01 = ld2(Wq + (size_t)128 * 128 + h * 16 + k);
    qv[i].x = q01.x + cap * w01.x;
    qv[i].y = q01.y + cap * w01.y;
  }

  // ---- score tiles: q(16x16) @ k^T, K = QK = 16 ----
#pragma unroll 1
  for (int pt = 0; pt < 32; ++pt) {
    const int p0 = pt * 16;
    const float* kr = kall + ((size_t)b * NPP + p0 + nl) * 128 + h * 16;
    v8f acc = v8_splat(0.f);
#pragma unroll
    for (int i = 0; i < 4; ++i) {  // FULL unroll: qv indices constant
      v2f bb = ld2(kr + i * 4 + half * 2);
      acc = wmma4(qv[i], bb, acc);
    }
#pragma unroll
    for (int r = 0; r < 8; ++r) {
      const int m = r + half * 8;
      int g = g0 + m; if (g >= NG) g = NG - 1;  // clamp (rows never stored)
      const int p = p0 + nl;
      float val;
      if (p >= NP) val = NEG_INF;
      else val = acc[r] * 0.25f + mask[((size_t)b * NG + g) * NP + p];
      s[m][p0 + nl] = val;
    }
  }
  __syncthreads();

  // ---- row softmax over p (512 padded, pads are -inf) ----
  for (int m = 0; m < 16; ++m) {
    float mx = NEG_INF;
    for (int j = lane; j < 512; j += 32) mx = fmaxf(mx, s[m][j]);
#pragma unroll
    for (int o = 16; o > 0; o >>= 1) mx = fmaxf(mx, __shfl_xor(mx, o, 32));
    float sm = 0.f;
    for (int j = lane; j < 512; j += 32) {
      const float sv = s[m][j];
      const float e = (sv == NEG_INF) ? 0.f : __expf(sv - mx);
      s[m][j] = e;
      sm += e;
    }
#pragma unroll
    for (int o = 16; o > 0; o >>= 1) sm += __shfl_xor(sm, o, 32);
    const float inv = 1.f / sm;
    for (int j = lane; j < 512; j += 32) s[m][j] *= inv;
  }
  __syncthreads();

  // ---- out = w(16x512) @ v(512x16), K = 512; vT rows are contiguous in p ---
  const float* vrow = vT + (((size_t)b * NH + h) * 16 + nl) * NPP;
  v8f acc = v8_splat(0.f);
#pragma unroll 8
  for (int k0 = 0; k0 < 512; k0 += 4) {
    const int k = k0 + half * 2;
    v2f a = ld2(&s[nl][k]);    // ds_load_b64
    v2f bb = ld2(vrow + k);    // global_load_b64 (w==0 on pads)
    acc = wmma4(a, bb, acc);
  }
  if (g0 + 16 <= NG) {
#pragma unroll
    for (int r = 0; r < 8; ++r)
      outc[((size_t)b * NG + g0 + r + half * 8) * 128 + h * 16 + nl] = acc[r];
  } else {
#pragma unroll
    for (int r = 0; r < 8; ++r) {
      const int g = g0 + r + half * 8;
      if (g < NG) outc[((size_t)b * NG + g) * 128 + h * 16 + nl] = acc[r];
    }
  }
}

// ---------------------------------------------------------------------------
// 4) mh = out_concat (16000x128) @ Wcomb (via WcombT); A in regs, 8 n-tiles
// ---------------------------------------------------------------------------
__global__ __launch_bounds__(32) void gemm_wcomb_kernel(
    const float* __restrict__ X, const float* __restrict__ WT,
    float* __restrict__ Y) {
  const int lane = threadIdx.x & 31;
  const int half = lane >> 4, nl = lane & 15;
  const int m0 = blockIdx.x * 16;
  const float* arow = X + (size_t)(m0 + nl) * 128;
  v2f areg[32];
#pragma unroll
  for (int i = 0; i < 32; ++i) areg[i] = ld2(arow + i * 4 + half * 2);
#pragma unroll 1
  for (int nt = 0; nt < 8; ++nt) {
    const int n0 = nt * 16;
    const float* wrow = WT + (size_t)(n0 + nl) * 128;
    v8f acc = v8_splat(0.f);
#pragma unroll
    for (int i = 0; i < 32; ++i) {  // FULL unroll: areg indices constant
      v2f bb = ld2(wrow + i * 4 + half * 2);
      acc = wmma4(areg[i], bb, acc);
    }
#pragma unroll
    for (int r = 0; r < 8; ++r)
      Y[(size_t)(m0 + r + half * 8) * 128 + n0 + nl] = acc[r];
  }
}

// ---------------------------------------------------------------------------
// 5) top-2 gating per token -> dense gates [T,8]
// ---------------------------------------------------------------------------
__global__ __launch_bounds__(128) void gate_kernel(
    const float* __restrict__ mh, const float* __restrict__ w_gate,
    float* __restrict__ gates) {
  const int t = blockIdx.x * 128 + threadIdx.x;
  if (t >= NT) return;
  float lg[8];
#pragma unroll
  for (int e = 0; e < 8; ++e) lg[e] = 0.f;
  const float* x = mh + (size_t)t * 128;
  for (int d = 0; d < 128; ++d) {
    const float xv = x[d];
    const float* wr = w_gate + (size_t)d * 8;
#pragma unroll
    for (int e = 0; e < 8; ++e) lg[e] += xv * wr[e];
  }
  int i0 = 0; float v0 = lg[0];
#pragma unroll
  for (int e = 1; e < 8; ++e) if (lg[e] > v0) { v0 = lg[e]; i0 = e; }
  int i1 = -1; float v1 = NEG_INF;
#pragma unroll
  for (int e = 0; e < 8; ++e) if (e != i0 && lg[e] > v1) { v1 = lg[e]; i1 = e; }
  const float ee = __expf(v1 - v0);
  const float g0 = 1.f / (1.f + ee);
  const float g1 = ee / (1.f + ee);
  float* gr = gates + (size_t)t * 8;
#pragma unroll
  for (int e = 0; e < 8; ++e) gr[e] = 0.f;
  gr[i0] = g0;
  gr[i1] = g1;
}

// deterministic importance[e] = sum_t gates[t,e]
__global__ __launch_bounds__(256) void importance_kernel(
    const float* __restrict__ gates, float* __restrict__ importance) {
  __shared__ float red[256];
  const int e = blockIdx.x, tid = threadIdx.x;
  float s = 0.f;
  for (int t = tid; t < NT; t += 256) s += gates[(size_t)t * 8 + e];
  red[tid] = s;
  __syncthreads();
  for (int o = 128; o > 0; o >>= 1) {
    if (tid < o) red[tid] += red[tid + o];
    __syncthreads();
  }
  if (tid == 0) importance[e] = red[0];
}

__global__ void loss_kernel(const float* __restrict__ imp,
                            float* __restrict__ out) {
  if (threadIdx.x == 0) {
    float s = 0.f;
#pragma unroll
    for (int e = 0; e < 8; ++e) s += imp[e];
    const float mean = s * 0.125f;
    float v = 0.f;
#pragma unroll
    for (int e = 0; e < 8; ++e) { const float d = imp[e] - mean; v += d * d; }
    v *= (1.f / 7.f);  // ddof=1
    out[(size_t)NB * NG * NP] = v / (mean * mean + 1e-10f);
  }
}

// ---------------------------------------------------------------------------
// 6) MoE: 4 waves per 16-token tile; skip inactive experts; added = mh + moe
//    Uses transposed W1T [E,FF,D], W2T [E,D,FF]; layer-1 A in registers.
// ---------------------------------------------------------------------------
__global__ __launch_bounds__(128) void moe_kernel(
    const float* __restrict__ mh, const float* __restrict__ gates,
    const float* __restrict__ W1T, const float* __restrict__ b1,
    const float* __restrict__ W2T, const float* __restrict__ b2,
    float* __restrict__ added) {
  __shared__ float Xs[16][128];
  __shared__ float Hs[16][512];
  __shared__ float Acc[16][128];
  __shared__ float gs[16][8];
  __shared__ int active[8];
  const int tid = threadIdx.x;
  const int wave = tid >> 5, lane = tid & 31;
  const int half = lane >> 4, nl = lane & 15;
  const size_t t0 = (size_t)blockIdx.x * 16;
  float* Xf = &Xs[0][0];
  float* Af = &Acc[0][0];
  float* Gf = &gs[0][0];
  for (int i = tid; i < 2048; i += 128) {
    Xf[i] = mh[t0 * 128 + i];
    Af[i] = 0.f;
  }
  if (tid < 128) Gf[tid] = gates[t0 * 8 + tid];
  __syncthreads();
  if (tid < 8) {
    int a = 0;
    for (int m = 0; m < 16; ++m) a |= (gs[m][tid] != 0.f);
    active[tid] = a;
  }
  // layer-1 A fragment in registers (invariant across experts and n-tiles)
  v2f xa[32];
#pragma unroll
  for (int i = 0; i < 32; ++i) xa[i] = ld2(&Xs[nl][i * 4 + half * 2]);
  __syncthreads();

#pragma unroll 1
  for (int e = 0; e < 8; ++e) {
    if (active[e]) {
      // layer 1: x(16x128) @ W1[e](128x512) + b1, relu -> Hs
#pragma unroll 1
      for (int nt = 0; nt < 8; ++nt) {
        const int n0 = wave * 128 + nt * 16;
        const float* wrow = W1T + ((size_t)e * NFF + n0 + nl) * 128;
        v8f acc = v8_splat(b1[(size_t)e * NFF + n0 + nl]);
#pragma unroll
        for (int i = 0; i < 32; ++i) {  // FULL unroll: xa indices constant
          v2f bb = ld2(wrow + i * 4 + half * 2);
          acc = wmma4(xa[i], bb, acc);
        }
#pragma unroll
        for (int r = 0; r < 8; ++r)
          Hs[r + half * 8][n0 + nl] = fmaxf(acc[r], 0.f);
      }
      __syncthreads();
      // layer 2: h(16x512) @ W2[e](512x128) + b2, scale by gate, accumulate
#pragma unroll 1
      for (int q = 0; q < 2; ++q) {
        const int n0 = (wave * 2 + q) * 16;
        const float* wrow = W2T + ((size_t)e * ND + n0 + nl) * 512;
        v8f acc = v8_splat(b2[(size_t)e * ND + n0 + nl]);
#pragma unroll 8
        for (int k0 = 0; k0 < 512; k0 += 4) {
          const int k = k0 + half * 2;
          v2f a = ld2(&Hs[nl][k]);   // ds_load_b64
          v2f bb = ld2(wrow + k);    // global_load_b64
          acc = wmma4(a, bb, acc);
        }
#pragma unroll
        for (int r = 0; r < 8; ++r) {
          const int m = r + half * 8;
          Acc[m][n0 + nl] += gs[m][e] * acc[r];
        }
      }
    }
    __syncthreads();
  }
  for (int i = tid; i < 2048; i += 128) added[t0 * 128 + i] = Xf[i] + Af[i];
}

// ---------------------------------------------------------------------------
// 7) InstanceNorm over group dim: one block per (b,d)
// ---------------------------------------------------------------------------
__global__ __launch_bounds__(128) void norm_kernel(
    const float* __restrict__ added, const float* __restrict__ nw,
    const float* __restrict__ nb, float* __restrict__ normed) {
  __shared__ float red[128];
  const int b = blockIdx.x >> 7, d = blockIdx.x & 127;
  const int tid = threadIdx.x;
  float x[4];
  float s = 0.f, ss = 0.f;
#pragma unroll
  for (int j = 0; j < 4; ++j) {
    const int g = tid + j * 128;
    const int gc = (g < NG) ? g : (NG - 1);     // clamped, unconditional load
    const float v = added[((size_t)b * NG + gc) * 128 + d];
    x[j] = v;
    const float vm = (g < NG) ? v : 0.f;        // select, no branch
    s += vm;
    ss += vm * vm;
  }
  red[tid] = s;
  __syncthreads();
  for (int o = 64; o > 0; o >>= 1) {
    if (tid < o) red[tid] += red[tid + o];
    __syncthreads();
  }
  const float mean = red[0] * (1.f / NG);
  __syncthreads();
  red[tid] = ss;
  __syncthreads();
  for (int o = 64; o > 0; o >>= 1) {
    if (tid < o) red[tid] += red[tid + o];
    __syncthreads();
  }
  const float var = red[0] * (1.f / NG) - mean * mean;
  const float scale = nw[d] * rsqrtf(var + 1e-5f);
  const float bias = nb[d];
#pragma unroll
  for (int j = 0; j < 4; ++j) {
    const int g = tid + j * 128;
    if (g < NG)
      normed[((size_t)b * NG + g) * 128 + d] = (x[j] - mean) * scale + bias;
  }
}

// ---------------------------------------------------------------------------
// 8) probs: one wave per (b, 16-row g-tile); A rows straight from global into
//    registers (reused across 32 p-tiles); WMMA score2, tanh-clip, softmax
// ---------------------------------------------------------------------------
__global__ __launch_bounds__(32) void probs_kernel(
    const float* __restrict__ normed, const float* __restrict__ enc,
    const float* __restrict__ mask, float* __restrict__ probs) {
  __shared__ float s[16][512];
  const int lane = threadIdx.x;
  const int half = lane >> 4, nl = lane & 15;
  const int gt = blockIdx.x, b = blockIdx.y;
  const int g0 = gt * 16;
  int ga = g0 + nl;
  if (ga >= NG) ga = NG - 1;  // clamp: duplicate rows never stored
  const float* nrow = normed + ((size_t)b * NG + ga) * 128;
  v2f areg[32];
#pragma unroll
  for (int i = 0; i < 32; ++i) areg[i] = ld2(nrow + i * 4 + half * 2);

#pragma unroll 1
  for (int pt = 0; pt < 32; ++pt) {
    const int p0 = pt * 16;
    int pc = p0 + nl; if (pc >= NP) pc = NP - 1;
    const float* erow = enc + ((size_t)b * NP + pc) * 128;
    v8f acc = v8_splat(0.f);
#pragma unroll
    for (int i = 0; i < 32; ++i) {  // FULL unroll: areg indices constant
      v2f bb = ld2(erow + i * 4 + half * 2);  // global_load_b64
      acc = wmma4(areg[i], bb, acc);
    }
#pragma unroll
    for (int r = 0; r < 8; ++r) {
      const int m = r + half * 8;
      int g = g0 + m; if (g >= NG) g = NG - 1;  // clamp
      const int p = p0 + nl;
      float val;
      if (p >= NP) val = NEG_INF;
      else val = 10.f * tanhf(acc[r] * INV_SQRT_EMB) +
                 mask[((size_t)b * NG + g) * NP + p];
      s[m][p0 + nl] = val;
    }
  }
  __syncthreads();

  for (int m = 0; m < 16; ++m) {
    float mx = NEG_INF;
    for (int j = lane; j < 512; j += 32) mx = fmaxf(mx, s[m][j]);
#pragma unroll
    for (int o = 16; o > 0; o >>= 1) mx = fmaxf(mx, __shfl_xor(mx, o, 32));
    float sm = 0.f;
    for (int j = lane; j < 512; j += 32) {
      const float sv = s[m][j];
      const float e = (sv == NEG_INF) ? 0.f : __expf(sv - mx);
      s[m][j] = e;
      sm += e;
    }
#pragma unroll
    for (int o = 16; o > 0; o >>= 1) sm += __shfl_xor(sm, o, 32);
    const float inv = 1.f / sm;
    for (int j = lane; j < 512; j += 32) s[m][j] *= inv;
  }
  __syncthreads();

  for (int m = 0; m < 16; ++m) {
    const int g = g0 + m;
    if (g >= NG) continue;
    for (int j = lane; j < NP; j += 32)
      probs[((size_t)b * NG + g) * NP + j] = s[m][j];
  }
}

// ---------------------------------------------------------------------------
extern "C" void kernel_launch(void* const* d_in, const int* in_sizes, int n_in,
                              void* d_out, int out_size, void* d_ws,
                              size_t ws_size, hipStream_t stream) {
  const float* graph    = (const float*)d_in[0];
  const float* capacity = (const float*)d_in[1];
  const float* mask     = (const float*)d_in[2];
  const float* enc      = (const float*)d_in[3];
  const float* Wq       = (const float*)d_in[4];
  const float* Wk       = (const float*)d_in[5];
  const float* Wv       = (const float*)d_in[6];
  const float* Wcomb    = (const float*)d_in[7];
  const float* w_gate   = (const float*)d_in[8];
  const float* W1       = (const float*)d_in[9];
  const float* b1       = (const float*)d_in[10];
  const float* W2       = (const float*)d_in[11];
  const float* b2       = (const float*)d_in[12];
  const float* nw       = (const float*)d_in[13];
  const float* nb       = (const float*)d_in[14];
  float* out = (float*)d_out;

  float* w = (float*)d_ws;
  const size_t SZK = (size_t)NB * NPP * 128;  // 2,097,152 (kall / vT)
  const size_t SZT = (size_t)NB * NG * 128;   // 2,048,000 (outc / mh)
  float* kall  = w;
  float* vT    = kall + SZK;
  float* outc  = vT + SZK;
  float* mh    = outc + SZT;
  float* gates = mh + SZT;                   // NT*8
  float* qbase = gates + (size_t)NT * 8;     // NB*128
  float* imp   = qbase + (size_t)NB * 128;   // 8
  float* WkT   = imp + 8;                    // 128*128
  float* WvT   = WkT + 128 * 128;
  float* WcombT = WvT + 128 * 128;
  float* W1T   = WcombT + 128 * 128;         // E*FF*D
  float* W2T   = W1T + (size_t)NE * NFF * ND;  // E*D*FF
  float* added  = outc;  // reuse: outc dead after Wcomb GEMM
  float* normed = kall;  // reuse: kall dead after attention

  // weight transposes (column-major B operands -> b64 fragment loads)
  transpose_batched<<<dim3(64, 1), 256, 0, stream>>>(Wk, WkT, 128, 128);
  transpose_batched<<<dim3(64, 1), 256, 0, stream>>>(Wv, WvT, 128, 128);
  transpose_batched<<<dim3(64, 1), 256, 0, stream>>>(Wcomb, WcombT, 128, 128);
  transpose_batched<<<dim3(256, NE), 256, 0, stream>>>(W1, W1T, 128, 512);
  transpose_batched<<<dim3(256, NE), 256, 0, stream>>>(W2, W2T, 512, 128);

  proj_kv_kernel<<<dim3(32, NB * 2), 32, 0, stream>>>(enc, WkT, WvT, kall, vT);
  qbase_kernel<<<NB, 128, 0, stream>>>(graph, Wq, qbase);
  attn_kernel<<<dim3(32, NH, NB), 32, 0, stream>>>(qbase, Wq, capacity, mask,
                                                   kall, vT, outc);
  gemm_wcomb_kernel<<<NT / 16, 32, 0, stream>>>(outc, WcombT, mh);
  gate_kernel<<<NT / 128, 128, 0, stream>>>(mh, w_gate, gates);
  importance_kernel<<<NE, 256, 0, stream>>>(gates, imp);
  loss_kernel<<<1, 32, 0, stream>>>(imp, out);
  moe_kernel<<<NT / 16, 128, 0, stream>>>(mh, gates, W1T, b1, W2T, b2, added);
  norm_kernel<<<NB * 128, 128, 0, stream>>>(added, nw, nb, normed);
  probs_kernel<<<dim3(32, NB), 32, 0, stream>>>(normed, enc, mask, out);
}